// MixingBlock_52003464020086
// MI455X (gfx1250) — compile-verified
//
#include <hip/hip_runtime.h>
#include <hip/hip_bf16.h>

typedef __attribute__((ext_vector_type(2))) float v2f;
typedef __attribute__((ext_vector_type(8))) float v8f;

#define B_DIM 4
#define R_DIM 8192
#define S_DIM 32
#define IN_DIM 180
#define A_DIM 1536
#define VSTRIDE 1544   // padded V row stride (dwords): 2*1544 % 64 == 16 -> conflict-free
#define WSTRIDE 34     // padded w row stride (dwords)

static __device__ __forceinline__ v8f wmma_f32_k4(v2f a, v2f b, v8f c) {
    return __builtin_amdgcn_wmma_f32_16x16x4_f32(false, a, false, b, (short)0, c,
                                                 false, false);
}

// ---------------------------------------------------------------------------
// Kernel 1: k_all[128,1536] = slot_flat[128,1536] @ Wk[1536,1536]
// ---------------------------------------------------------------------------
__global__ __launch_bounds__(256) void k_gemm_kernel(
    const float* __restrict__ slot, const float* __restrict__ Wk,
    float* __restrict__ k_ws) {
    const int lane = threadIdx.x & 31;
    const int wave = threadIdx.x >> 5;
    const int tile = blockIdx.x * 8 + wave;   // 0..767
    const int tm = tile & 7;                  // 8 row tiles (128 rows)
    const int tn = tile >> 3;                 // 96 col tiles
    const int l16 = lane & 15;
    const int halfK = (lane >> 4) * 2;        // 0 or 2
    const int mrow = tm * 16 + l16;
    const int ncol = tn * 16 + l16;

    v8f acc = {};
    for (int kk = 0; kk < A_DIM / 4; ++kk) {
        const int k0 = kk * 4 + halfK;
        v2f a = *(const v2f*)(slot + (size_t)mrow * A_DIM + k0);
        v2f b;
        b.x = Wk[(size_t)(k0 + 0) * A_DIM + ncol];
        b.y = Wk[(size_t)(k0 + 1) * A_DIM + ncol];
        acc = wmma_f32_k4(a, b, acc);
    }
    const int rbase = tm * 16 + (lane >> 4) * 8;
#pragma unroll
    for (int j = 0; j < 8; ++j)
        k_ws[(size_t)(rbase + j) * A_DIM + ncol] = acc[j];
}

// ---------------------------------------------------------------------------
// Kernel 2: P[b][180,32] = Wq[180,1536] @ k[b][32,1536]^T
// ---------------------------------------------------------------------------
__global__ __launch_bounds__(256) void p_gemm_kernel(
    const float* __restrict__ Wq, const float* __restrict__ k_ws,
    float* __restrict__ P_ws) {
    const int lane = threadIdx.x & 31;
    const int wave = threadIdx.x >> 5;
    const int tile = blockIdx.x * 8 + wave;   // 0..95
    const int tn = tile & 1;
    const int tm = (tile >> 1) % 12;
    const int b  = tile / 24;
    const int l16 = lane & 15;
    const int halfK = (lane >> 4) * 2;
    const int mrow = tm * 16 + l16;
    const int mrow_c = (mrow < IN_DIM) ? mrow : (IN_DIM - 1);  // clamp pad rows
    const int s = tn * 16 + l16;

    v8f acc = {};
    for (int kk = 0; kk < A_DIM / 4; ++kk) {
        const int k0 = kk * 4 + halfK;
        v2f a = *(const v2f*)(Wq + (size_t)mrow_c * A_DIM + k0);
        v2f bf = *(const v2f*)(k_ws + (size_t)(b * S_DIM + s) * A_DIM + k0);
        acc = wmma_f32_k4(a, bf, acc);
    }
    const int rbase = tm * 16 + (lane >> 4) * 8;
#pragma unroll
    for (int j = 0; j < 8; ++j) {
        const int i = rbase + j;
        if (i < IN_DIM) P_ws[((size_t)b * IN_DIM + i) * S_DIM + s] = acc[j];
    }
}

// ---------------------------------------------------------------------------
// Kernel 3: per workgroup: batch b, 128 rows (8 waves x 16 rows).
//   async-stage V into LDS, overlap with dots = x@P; softmax; s = w@V.
// ---------------------------------------------------------------------------
__global__ __launch_bounds__(256) void attn_kernel(
    const float* __restrict__ x, const float* __restrict__ slot,
    const float* __restrict__ P_ws, float* __restrict__ s_out,
    float* __restrict__ w_out) {
    extern __shared__ float smem[];
    float* Vlds = smem;                         // [32][VSTRIDE]
    float* wlds = smem + S_DIM * VSTRIDE;       // [8][16][WSTRIDE]

    const int lane = threadIdx.x & 31;
    const int wave = threadIdx.x >> 5;
    const int b    = blockIdx.x >> 6;           // 4 batches
    const int rblk = blockIdx.x & 63;           // 64 blocks of 128 rows

    // --- issue async global->LDS stage of V; overlaps with the dots GEMM ---
    // dynamic LDS starts at offset 0 (no static LDS in this kernel)
    {
        const float* Vg = slot + (size_t)b * S_DIM * A_DIM;
        for (int idx = threadIdx.x; idx < S_DIM * (A_DIM / 4); idx += 256) {
            const int row  = idx / (A_DIM / 4);
            const int col4 = idx % (A_DIM / 4);
            const float* gsrc = Vg + (size_t)row * A_DIM + col4 * 4;
            const unsigned lds_byte =
                (unsigned)((row * VSTRIDE + col4 * 4) * sizeof(float));
            asm volatile("global_load_async_to_lds_b128 %0, %1, off"
                         :
                         : "v"(lds_byte), "v"(gsrc)
                         : "memory");
        }
    }

    const int l16 = lane & 15;
    const int halfK = (lane >> 4) * 2;
    const int row0 = rblk * 128 + wave * 16;
    const int mrow = row0 + l16;
    const float* xrow = x + ((size_t)b * R_DIM + mrow) * IN_DIM;
    const float* Pb = P_ws + (size_t)b * IN_DIM * S_DIM;

    // ---- dots = x @ P : K = 180 = 45 * 4, two N tiles (32 slots) ----
    v8f acc0 = {}, acc1 = {};
    for (int kk = 0; kk < IN_DIM / 4; ++kk) {
        const int k0 = kk * 4 + halfK;
        v2f a = __builtin_nontemporal_load((const v2f*)(xrow + k0));
        v2f b0, b1;
        b0.x = Pb[(k0 + 0) * S_DIM + l16];
        b0.y = Pb[(k0 + 1) * S_DIM + l16];
        b1.x = Pb[(k0 + 0) * S_DIM + 16 + l16];
        b1.y = Pb[(k0 + 1) * S_DIM + 16 + l16];
        acc0 = wmma_f32_k4(a, b0, acc0);
        acc1 = wmma_f32_k4(a, b1, acc1);
    }
    float* wtile = wlds + wave * 16 * WSTRIDE;
    {
        const int rb = (lane >> 4) * 8;
#pragma unroll
        for (int j = 0; j < 8; ++j) {
            wtile[(rb + j) * WSTRIDE + l16]      = acc0[j];
            wtile[(rb + j) * WSTRIDE + 16 + l16] = acc1[j];
        }
    }
    // V stage must be complete and visible to all waves before phase 4
    asm volatile("s_wait_asynccnt 0x0" ::: "memory");
    __syncthreads();

    // ---- softmax over the 32 slots: lanes 0..15 each own one row ----
    const float scale = 0.02551551815f;  // 1536^-0.5
    if (lane < 16) {
        float vals[S_DIM];
        float m = -3.4e38f;
#pragma unroll
        for (int s = 0; s < S_DIM; ++s) {
            float v = wtile[lane * WSTRIDE + s] * scale;
            vals[s] = v;
            m = fmaxf(m, v);
        }
        float sum = 0.f;
#pragma unroll
        for (int s = 0; s < S_DIM; ++s) {
            float e = __expf(vals[s] - m);
            vals[s] = e;
            sum += e;
        }
        const float inv = 1.0f / sum;
        float* wrow = w_out + ((size_t)b * R_DIM + row0 + lane) * S_DIM;
#pragma unroll
        for (int s = 0; s < S_DIM; ++s) {
            float wv = vals[s] * inv;
            wtile[lane * WSTRIDE + s] = wv;
            __builtin_nontemporal_store(wv, wrow + s);
        }
    }
    __syncthreads();

    // ---- s = w[16,32] @ V[32,1536] : 8 K-steps, 2 N-tiles per iteration ----
    v2f afrag[8];
#pragma unroll
    for (int kk = 0; kk < 8; ++kk) {
        const int k0 = kk * 4 + halfK;
        afrag[kk].x = wtile[l16 * WSTRIDE + k0];
        afrag[kk].y = wtile[l16 * WSTRIDE + k0 + 1];
    }
    float* orow = s_out + ((size_t)b * R_DIM + row0) * A_DIM;
    const int rb = (lane >> 4) * 8;
    for (int it = 0; it < (A_DIM / 16) / 2; ++it) {
        const int n0 = (2 * it) * 16 + l16;
        const int n1 = n0 + 16;
        v8f accA = {}, accB = {};
#pragma unroll
        for (int kk = 0; kk < 8; ++kk) {
            const int k0 = kk * 4 + halfK;
            v2f bfA, bfB;
            bfA.x = Vlds[(k0 + 0) * VSTRIDE + n0];
            bfA.y = Vlds[(k0 + 1) * VSTRIDE + n0];
            bfB.x = Vlds[(k0 + 0) * VSTRIDE + n1];
            bfB.y = Vlds[(k0 + 1) * VSTRIDE + n1];
            accA = wmma_f32_k4(afrag[kk], bfA, accA);
            accB = wmma_f32_k4(afrag[kk], bfB, accB);
        }
#pragma unroll
        for (int j = 0; j < 8; ++j) {
            __builtin_nontemporal_store(accA[j], orow + (size_t)(rb + j) * A_DIM + n0);
            __builtin_nontemporal_store(accB[j], orow + (size_t)(rb + j) * A_DIM + n1);
        }
    }
}

extern "C" void kernel_launch(void* const* d_in, const int* in_sizes, int n_in,
                              void* d_out, int out_size, void* d_ws, size_t ws_size,
                              hipStream_t stream) {
    (void)in_sizes; (void)n_in; (void)out_size; (void)ws_size;
    const float* x    = (const float*)d_in[0];
    const float* slot = (const float*)d_in[1];
    const float* Wq   = (const float*)d_in[2];
    const float* Wk   = (const float*)d_in[3];
    float* s_out = (float*)d_out;                                   // [4,8192,1536]
    float* w_out = s_out + (size_t)B_DIM * R_DIM * A_DIM;           // [4,8192,32]
    float* k_ws  = (float*)d_ws;                                    // [128,1536]
    float* P_ws  = k_ws + (size_t)B_DIM * S_DIM * A_DIM;            // [4,180,32]

    k_gemm_kernel<<<96, 256, 0, stream>>>(slot, Wk, k_ws);
    p_gemm_kernel<<<12, 256, 0, stream>>>(Wq, k_ws, P_ws);
    const size_t shmem =
        (size_t)(S_DIM * VSTRIDE + 8 * 16 * WSTRIDE) * sizeof(float);
    attn_kernel<<<B_DIM * 64, 256, shmem, stream>>>(x, slot, P_ws, s_out, w_out);
}